// ONLSTM_47880295416039
// MI455X (gfx1250) — compile-verified
//
#include <hip/hip_runtime.h>
#include <math.h>

// ---------------- problem constants ----------------
#define BQ     32          // batch
#define TT     512         // timesteps
#define DD     1024        // input dim
#define UU     1024        // units
#define LL     16          // levels
#define CC     64          // chunk
#define GG     4128        // 4*UU + 2*LL
#define KTOT   2048        // concatenated K = D + U
#define KSPLIT 8           // K split across blocks
#define KSLICE (KTOT / KSPLIT)   // 256
#define NTILES (GG / 16)         // 258

typedef __attribute__((ext_vector_type(2))) float v2f;
typedef __attribute__((ext_vector_type(8))) float v8f;

// ---------------- zero init for h, c state ----------------
__global__ void onlstm_zero_kernel(float* p, int n) {
    int i = blockIdx.x * blockDim.x + threadIdx.x;
    if (i < n) p[i] = 0.0f;
}

// ---------------- per-step GEMM: zp[kz] += [x_t | h] @ [W ; R] ----------------
// grid = (NTILES, KSPLIT), block = 32 (one wave). Each wave computes two
// 16x16 output tiles (M rows 0-15 and 16-31 of the 32-row batch) for one
// 16-column N tile and one K slice, using V_WMMA_F32_16X16X4_F32.
__global__ void __launch_bounds__(32)
onlstm_step_gemm(const float* __restrict__ X,   // (B, T, D)
                 const float* __restrict__ W,   // (D, G)
                 const float* __restrict__ R,   // (U, G)
                 const float* __restrict__ H,   // (B, U)  h_{t-1}
                 float* __restrict__ zp,        // (KSPLIT, B, G) partial sums
                 int t) {
    const int ntile = blockIdx.x;          // 0..257
    const int kz    = blockIdx.y;          // 0..7
    const int lane  = threadIdx.x;         // 0..31
    const int laneLow = lane & 15;
    const int hi      = lane >> 4;         // 0 for lanes 0-15, 1 for 16-31
    const int n0      = ntile * 16;

    // Select source for this K slice: kz 0..3 -> x_t @ W, kz 4..7 -> h @ R.
    const float* A;       // base of A matrix rows
    size_t aStride;       // row stride of A
    size_t aOff;          // per-row starting offset for this slice
    const float* Bm;      // B matrix rows for this slice (row k is contiguous)
    const int kglob = kz * KSLICE;
    if (kglob < DD) {
        A = X; aStride = (size_t)TT * DD; aOff = (size_t)t * DD + kglob;
        Bm = W + (size_t)kglob * GG;
    } else {
        A = H; aStride = UU; aOff = (size_t)(kglob - DD);
        Bm = R + (size_t)(kglob - DD) * GG;
    }

    v8f acc0 = {};   // rows 0..15  (M tile 0)
    v8f acc1 = {};   // rows 16..31 (M tile 1)

    // A 16x4 layout: lane holds A[m = laneLow][k + 2*hi + {0,1}]  -> float2 load
    // B 4x16 layout: VGPR0 = row k (lanes 0-15) / row k+2 (lanes 16-31), N = laneLow
    const float* a0base = A + (size_t)laneLow * aStride + aOff + 2 * hi;
    const float* a1base = A + (size_t)(laneLow + 16) * aStride + aOff + 2 * hi;
    const float* bbase  = Bm + (size_t)(2 * hi) * GG + n0 + laneLow;

    #pragma unroll 4
    for (int kk = 0; kk < KSLICE; kk += 4) {
        v2f a0 = *(const v2f*)(a0base + kk);
        v2f a1 = *(const v2f*)(a1base + kk);
        v2f bv;
        bv.x = bbase[(size_t)kk * GG];
        bv.y = bbase[(size_t)(kk + 1) * GG];
        acc0 = __builtin_amdgcn_wmma_f32_16x16x4_f32(
            false, a0, false, bv, (short)0, acc0, false, false);
        acc1 = __builtin_amdgcn_wmma_f32_16x16x4_f32(
            false, a1, false, bv, (short)0, acc1, false, false);
    }

    // C/D layout: VGPR j -> M = j (lanes 0-15) or M = j + 8 (lanes 16-31)
    float* out = zp + (size_t)kz * BQ * GG;
    #pragma unroll
    for (int j = 0; j < 8; ++j) {
        int m = j + 8 * hi;
        out[(size_t)m * GG + n0 + laneLow]        = acc0[j];
        out[(size_t)(16 + m) * GG + n0 + laneLow] = acc1[j];
    }
}

__device__ __forceinline__ float sigf(float x) {
    return 1.0f / (1.0f + __expf(-x));
}

// ---------------- per-step gates / state update ----------------
// grid = B, block = 256. Reduces the KSPLIT partials + bias into LDS,
// computes cumulative-softmax master gates, then the ON-LSTM cell update.
__global__ void __launch_bounds__(256)
onlstm_step_gates(const float* __restrict__ zp,    // (KSPLIT, B, G)
                  const float* __restrict__ bias,  // (G,)
                  float* __restrict__ cbuf,        // (B, U)
                  float* __restrict__ hbuf,        // (B, U)
                  float* __restrict__ hseq,        // (B, T, U)
                  int t) {
    __shared__ float zsh[GG];
    __shared__ float cumF[LL];
    __shared__ float cumI[LL];

    const int b   = blockIdx.x;
    const int tid = threadIdx.x;

    // reduce K-split partials + bias
    for (int j = tid; j < GG; j += 256) {
        float s = bias[j];
        #pragma unroll
        for (int p = 0; p < KSPLIT; ++p)
            s += zp[((size_t)p * BQ + b) * GG + j];
        zsh[j] = s;
    }
    __syncthreads();

    // master gates: cumsoftmax l2r on z[0:16], r2l (suffix) on z[16:32]
    if (tid == 0) {
        float mx = -3.4e38f;
        for (int l = 0; l < LL; ++l) mx = fmaxf(mx, zsh[l]);
        float e[LL], se = 0.0f;
        for (int l = 0; l < LL; ++l) { e[l] = __expf(zsh[l] - mx); se += e[l]; }
        float inv = 1.0f / se, run = 0.0f;
        for (int l = 0; l < LL; ++l) { run += e[l] * inv; cumF[l] = run; }

        mx = -3.4e38f;
        for (int l = 0; l < LL; ++l) mx = fmaxf(mx, zsh[LL + l]);
        se = 0.0f;
        for (int l = 0; l < LL; ++l) { e[l] = __expf(zsh[LL + l] - mx); se += e[l]; }
        inv = 1.0f / se; run = 0.0f;
        for (int l = LL - 1; l >= 0; --l) { run += e[l] * inv; cumI[l] = run; }
    }
    __syncthreads();

    // cell update: u = l*CC + ci
    for (int u = tid; u < UU; u += 256) {
        int l  = u >> 6;
        int ci = u & (CC - 1);
        float fm = cumF[l];
        float im = cumI[l];
        const int base = 2 * LL;  // gates start at z[32]
        float fg  = sigf(zsh[base + (l           ) * CC + ci]);
        float ig  = sigf(zsh[base + (LL     + l  ) * CC + ci]);
        float og  = sigf(zsh[base + (2 * LL + l  ) * CC + ci]);
        float cin = tanhf(zsh[base + (3 * LL + l ) * CC + ci]);
        float cp  = cbuf[(size_t)b * UU + u];
        float ov  = fm * im;
        float cn  = ov * (fg * cp + ig * cin) + (fm - ov) * cp + (im - ov) * cin;
        float hn  = og * tanhf(cn);
        cbuf[(size_t)b * UU + u] = cn;
        hbuf[(size_t)b * UU + u] = hn;
        hseq[((size_t)b * TT + t) * UU + u] = hn;
    }
}

extern "C" void kernel_launch(void* const* d_in, const int* in_sizes, int n_in,
                              void* d_out, int out_size, void* d_ws, size_t ws_size,
                              hipStream_t stream) {
    const float* x    = (const float*)d_in[0];   // (B, T, D)
    const float* W    = (const float*)d_in[1];   // (D, G)
    const float* R    = (const float*)d_in[2];   // (U, G)
    const float* bias = (const float*)d_in[3];   // (G,)
    float* hseq = (float*)d_out;                 // (B, T, U)

    // workspace layout (floats): zp | h | c
    float* zp   = (float*)d_ws;                          // KSPLIT*B*G
    float* hbuf = zp + (size_t)KSPLIT * BQ * GG;         // B*U
    float* cbuf = hbuf + (size_t)BQ * UU;                // B*U

    // zero initial state h0, c0
    {
        int n = 2 * BQ * UU;
        onlstm_zero_kernel<<<(n + 255) / 256, 256, 0, stream>>>(hbuf, n);
    }

    dim3 ggrid(NTILES, KSPLIT);
    for (int t = 0; t < TT; ++t) {
        onlstm_step_gemm<<<ggrid, 32, 0, stream>>>(x, W, R, hbuf, zp, t);
        onlstm_step_gates<<<BQ, 256, 0, stream>>>(zp, bias, cbuf, hbuf, hseq, t);
    }
}